// sigOut_27685359190619
// MI455X (gfx1250) — compile-verified
//
#include <hip/hip_runtime.h>
#include <cstdint>

// ---------------------------------------------------------------------------
// Sigmoid-output recurrence scan over last axis (T=64) of [4096,1024,64] f32.
//   u_t = TAU*u_{t-1}*(1 - o_{t-1}) + x_t ;  o_t = sigmoid(u_t)
// Memory-bound (2 GiB traffic @ 23.3 TB/s ~ 86 us floor): stream through LDS
// with CDNA5 async global<->LDS DMA (ASYNCcnt), scan per-lane rows from LDS.
// Async copies are fully coalesced (512 B per instruction per wave) and use
// non-temporal cache policy since the stream has zero reuse (>> 192 MB L2).
// ---------------------------------------------------------------------------

#define TAU_F 0.25f
#define T_LEN 64
#define ROWS_PER_WAVE 32
#define PAD_ROW 68                 // 64 + 4 floats pad: 16B-aligned rows, 2-way max bank conflict
#define WAVES_PER_BLOCK 4
#define THREADS_PER_BLOCK (WAVES_PER_BLOCK * 32)
#define F4_PER_CHUNK ((ROWS_PER_WAVE * T_LEN) / 4)   // 512 float4 per wave chunk
#define COPY_ITERS (F4_PER_CHUNK / 32)               // 16 async b128 ops per wave

#define CPOL_TH_NT 1               // non-temporal hint (TH bits [2:0] of cpol)

// Builtin-parameter types (probe-confirmed in round 1 diagnostics):
//   param0: 'int __vector(4) __device__ *'  (AS1 global, int4 pointee)
typedef int v4i __attribute__((vector_size(16)));
typedef __attribute__((address_space(1))) v4i* g_v4i_p;
typedef __attribute__((address_space(3))) v4i* l_v4i_p;

// Build AS pointers by value (global AS1 == flat value identity; LDS AS3 ==
// low 32 bits of the flat shared-aperture address, per aperture truncation).
__device__ __forceinline__ g_v4i_p to_g(const void* p) {
    return (g_v4i_p)(uintptr_t)p;
}
__device__ __forceinline__ l_v4i_p to_l(const void* p) {
    return (l_v4i_p)(uint32_t)(uintptr_t)p;
}

// ---- CDNA5 async global->LDS copy (ASYNCcnt-tracked), 16B per lane ----
__device__ __forceinline__ void async_g2l_b128(const float* gsrc, float* ldst) {
#if __has_builtin(__builtin_amdgcn_global_load_async_to_lds_b128)
    __builtin_amdgcn_global_load_async_to_lds_b128(to_g(gsrc), to_l(ldst),
                                                   /*imm_off=*/0, CPOL_TH_NT);
#else
    uint32_t loff = (uint32_t)(uintptr_t)ldst;
    asm volatile("global_load_async_to_lds_b128 %0, %1, off"
                 :: "v"(loff), "v"(gsrc) : "memory");
#endif
}

// ---- CDNA5 async LDS->global copy (ASYNCcnt-tracked), 16B per lane ----
__device__ __forceinline__ void async_l2g_b128(float* gdst, const float* lsrc) {
#if __has_builtin(__builtin_amdgcn_global_store_async_from_lds_b128)
    __builtin_amdgcn_global_store_async_from_lds_b128(to_g(gdst), to_l(lsrc),
                                                      /*imm_off=*/0, CPOL_TH_NT);
#else
    uint32_t loff = (uint32_t)(uintptr_t)lsrc;
    asm volatile("global_store_async_from_lds_b128 %0, %1, off"
                 :: "v"(gdst), "v"(loff) : "memory");
#endif
}

// Pure-asm waits: "memory" clobber guarantees the compiler cannot move LDS
// accesses across the hardware wait (s_wait_* asm syntax verified round 2).
__device__ __forceinline__ void wait_async0() {
    asm volatile("s_wait_asynccnt 0" ::: "memory");
}
__device__ __forceinline__ void wait_ds0() {
    asm volatile("s_wait_dscnt 0" ::: "memory");
}

// sigmoid(u) = rcp(1 + exp2(-u * log2(e)))  -> v_exp_f32 + v_rcp_f32
__device__ __forceinline__ float fast_sigmoid(float u) {
#if __has_builtin(__builtin_amdgcn_exp2f) && __has_builtin(__builtin_amdgcn_rcpf)
    float e = __builtin_amdgcn_exp2f(u * -1.4426950408889634f);
    return __builtin_amdgcn_rcpf(1.0f + e);
#else
    return 1.0f / (1.0f + exp2f(u * -1.4426950408889634f));
#endif
}

__global__ void __launch_bounds__(THREADS_PER_BLOCK)
sigOut_scan_kernel(const float* __restrict__ x, float* __restrict__ out,
                   long long nrows) {
    __shared__ __align__(16) float lds[WAVES_PER_BLOCK][ROWS_PER_WAVE * PAD_ROW];

    const int lane = threadIdx.x & 31;
    const int wave = threadIdx.x >> 5;
    const long long waveIdx = (long long)blockIdx.x * WAVES_PER_BLOCK + wave;
    const long long rowBase = waveIdx * ROWS_PER_WAVE;
    if (rowBase >= nrows) return;  // wave-uniform exit

    if (rowBase + ROWS_PER_WAVE <= nrows) {
        // ---------------- fast path: full 32-row chunk (8 KB) ----------------
        float* wbuf = &lds[wave][0];
        const long long gbase = rowBase * T_LEN;  // in floats

        // Phase 1: coalesced async DMA global -> LDS (512 B per instruction).
#pragma unroll
        for (int j = 0; j < COPY_ITERS; ++j) {
            int idx = j * 32 + lane;       // float4 index within chunk
            int r = idx >> 4;              // row within chunk
            int s = idx & 15;              // float4 segment within row
            async_g2l_b128(x + gbase + (long long)idx * 4,
                           &wbuf[r * PAD_ROW + s * 4]);
        }
        wait_async0();

        // Phase 2: per-lane sequential scan of row `lane`, in place in LDS.
        float4* rowv = (float4*)&wbuf[lane * PAD_ROW];  // 16B-aligned (272B stride)
        float u = 0.0f, o = 0.0f;
#pragma unroll
        for (int k = 0; k < T_LEN / 4; ++k) {
            float4 v = rowv[k];            // ds_load_b128
            u = TAU_F * u * (1.0f - o) + v.x;  o = fast_sigmoid(u);  v.x = o;
            u = TAU_F * u * (1.0f - o) + v.y;  o = fast_sigmoid(u);  v.y = o;
            u = TAU_F * u * (1.0f - o) + v.z;  o = fast_sigmoid(u);  v.z = o;
            u = TAU_F * u * (1.0f - o) + v.w;  o = fast_sigmoid(u);  v.w = o;
            rowv[k] = v;                   // ds_store_b128
        }

        // Phase 3: make LDS results visible to DMA, async store LDS -> global.
        // (No trailing asynccnt wait: S_ENDPGM performs implicit S_WAIT_IDLE.)
        wait_ds0();
#pragma unroll
        for (int j = 0; j < COPY_ITERS; ++j) {
            int idx = j * 32 + lane;
            int r = idx >> 4;
            int s = idx & 15;
            async_l2g_b128(out + gbase + (long long)idx * 4,
                           &wbuf[r * PAD_ROW + s * 4]);
        }
    } else {
        // ---------------- tail path: plain per-lane scalar scan --------------
        long long row = rowBase + lane;
        if (row < nrows) {
            const float* xr = x + row * T_LEN;
            float* orow = out + row * T_LEN;
            float u = 0.0f, o = 0.0f;
#pragma unroll 4
            for (int t = 0; t < T_LEN; ++t) {
                u = TAU_F * u * (1.0f - o) + xr[t];
                o = fast_sigmoid(u);
                orow[t] = o;
            }
        }
    }
}

extern "C" void kernel_launch(void* const* d_in, const int* in_sizes, int n_in,
                              void* d_out, int out_size, void* d_ws, size_t ws_size,
                              hipStream_t stream) {
    (void)n_in; (void)out_size; (void)d_ws; (void)ws_size;
    const float* x = (const float*)d_in[0];
    float* out = (float*)d_out;

    const long long n_elems = (long long)in_sizes[0];      // 4096*1024*64
    const long long nrows   = n_elems / T_LEN;             // 4,194,304
    const long long rows_per_block = (long long)ROWS_PER_WAVE * WAVES_PER_BLOCK; // 128
    const long long nblocks = (nrows + rows_per_block - 1) / rows_per_block;     // 32768

    sigOut_scan_kernel<<<dim3((unsigned)nblocks), dim3(THREADS_PER_BLOCK), 0, stream>>>(
        x, out, nrows);
}